// QAPPolicy_40475771798065
// MI455X (gfx1250) — compile-verified
//
#include <hip/hip_runtime.h>
#include <math.h>

namespace {

constexpr int kMB = 512, kT = 256, kNP1 = 201, kD = 4, kK = 5, kCTX = 8;
constexpr int kNTiles = 13;   // ceil(201/16) n-tiles
constexpr int kTTile  = 16;   // t rows per wave

typedef __attribute__((ext_vector_type(2))) float v2f;
typedef __attribute__((ext_vector_type(8))) float v8f;

__device__ __forceinline__ int lane_id() { return (int)(threadIdx.x & 31u); }

// wave32 xor-shuffle via ds_bpermute (byte index)
__device__ __forceinline__ float shflx_f(float v, int m) {
  return __int_as_float(
      __builtin_amdgcn_ds_bpermute((lane_id() ^ m) << 2, __float_as_int(v)));
}
__device__ __forceinline__ int shflx_i(int v, int m) {
  return __builtin_amdgcn_ds_bpermute((lane_id() ^ m) << 2, v);
}

// ---------------------------------------------------------------------------
// Kernel 1: sequential scan over T per batch element (1 thread per batch).
// Emits per-(b,t) pre-step state: current node, used capacity, visited bits.
// ---------------------------------------------------------------------------
__global__ void scan_kernel(const float* __restrict__ demands,
                            const int* __restrict__ actions,
                            int* __restrict__ cur_ws,        // [MB*T]
                            float* __restrict__ used_ws,     // [MB*T]
                            unsigned* __restrict__ vis_ws) { // [MB*T*7]
  const int b = blockIdx.x * blockDim.x + threadIdx.x;
  if (b >= kMB) return;
  int cur = 0;
  float used = 0.f;
  unsigned vis[7] = {0u, 0u, 0u, 0u, 0u, 0u, 0u};
  for (int t = 0; t < kT; ++t) {
    const size_t bt = (size_t)b * kT + t;
    cur_ws[bt] = cur;
    used_ws[bt] = used;
#pragma unroll
    for (int q = 0; q < 7; ++q) vis_ws[bt * 7 + q] = vis[q];
    const int a = actions[bt];
    const float dem = demands[(size_t)b * kNP1 + a];
    used = (a == 0) ? 0.f : (used + dem);
    vis[a >> 5] |= (1u << (a & 31));
    vis[0] &= ~1u;  // depot never marked visited
    cur = a;
  }
}

// ---------------------------------------------------------------------------
// Kernel 2: interference[b,n] = sum_k dot(psi[b,n], psi[b,knn[b,n,k]])
// ---------------------------------------------------------------------------
__global__ void interf_kernel(const float* __restrict__ psi,
                              const int* __restrict__ knn,
                              float* __restrict__ interf) {
  const int idx = blockIdx.x * blockDim.x + threadIdx.x;  // b*NP1 + n
  if (idx >= kMB * kNP1) return;
  const int b = idx / kNP1;
  const float* p = psi + (size_t)idx * kD;
  const float p0 = p[0], p1 = p[1], p2 = p[2], p3 = p[3];
  const int* kk = knn + (size_t)idx * kK;
  float acc = 0.f;
#pragma unroll
  for (int k = 0; k < kK; ++k) {
    const float* q = psi + ((size_t)b * kNP1 + kk[k]) * kD;
    acc += p0 * q[0] + p1 * q[1] + p2 * q[2] + p3 * q[3];
  }
  interf[idx] = acc;
}

// ---------------------------------------------------------------------------
// Kernel 3: fused scoring + mask + log-softmax + gather + entropy.
// Block = 256 threads = 8 waves; each wave owns one 16-row t-tile of one b.
// context_scores via V_WMMA_F32_16X16X4_F32 (A=query 16x4, B=psi^T 4x16).
// ---------------------------------------------------------------------------
__global__ __launch_bounds__(256, 1) void policy_main(
    const float* __restrict__ demands, const float* __restrict__ coords,
    const float* __restrict__ psi, const float* __restrict__ Wq_w,
    const float* __restrict__ Wq_b, const float* __restrict__ lam_p,
    const float* __restrict__ mu_p, const float* __restrict__ nu_p,
    const int* __restrict__ actions, const int* __restrict__ cap_p,
    const int* __restrict__ cur_ws, const float* __restrict__ used_ws,
    const unsigned* __restrict__ vis_ws, const float* __restrict__ interf_ws,
    float* __restrict__ out_lp, float* __restrict__ ent_ws) {
  const int b = blockIdx.x >> 1;
  const int halfB = blockIdx.x & 1;
  const int w = threadIdx.x >> 5;
  const int lane = threadIdx.x & 31;
  const int tl = lane & 15;      // column / t-within-tile index
  const int hi16 = lane >> 4;    // which 16-lane half
  const int ttile = halfB * 8 + w;
  const int tbase = ttile * kTTile;

  const float cap = fmaxf((float)cap_p[0], 1e-8f);
  const float lam = fminf(fmaxf(lam_p[0], -2.f), 3.f);
  const float mu  = fminf(fmaxf(mu_p[0], 0.f), 20.f);
  const float nu  = fminf(fmaxf(nu_p[0], -2.f), 3.f);

  __builtin_prefetch(psi + (size_t)b * kNP1 * kD, 0, 1);

  // Per-wave LDS staging of per-t state for the 16 rows of this tile.
  __shared__ int      s_cur[8][16];
  __shared__ float    s_rem[8][16];
  __shared__ int      s_act[8][16];
  __shared__ unsigned s_vis[8][16][7];

  if (lane < 16) {
    const int t = tbase + lane;
    const size_t bt = (size_t)b * kT + t;
    s_cur[w][lane] = cur_ws[bt];
    s_rem[w][lane] = cap - used_ws[bt];
    s_act[w][lane] = actions[bt];
#pragma unroll
    for (int q = 0; q < 7; ++q) s_vis[w][lane][q] = vis_ws[bt * 7 + q];
  }
  __syncthreads();

  // Uniform weight loads (scalar path)
  float wqw[kD][kCTX], wqb[kD];
#pragma unroll
  for (int d0 = 0; d0 < kD; ++d0) {
    wqb[d0] = Wq_b[d0];
#pragma unroll
    for (int j = 0; j < kCTX; ++j) wqw[d0][j] = Wq_w[d0 * kCTX + j];
  }

  // Per-row data for this half's 8 t rows (row tr = r + hi16*8)
  float curx[8], cury[8], rem8[8];
  int act8[8];
  bool dep8[8];
#pragma unroll
  for (int r = 0; r < 8; ++r) {
    const int tr = r + hi16 * 8;
    const int cur = s_cur[w][tr];
    curx[r] = coords[((size_t)b * kNP1 + cur) * 2 + 0];
    cury[r] = coords[((size_t)b * kNP1 + cur) * 2 + 1];
    dep8[r] = (cur == 0);
    rem8[r] = s_rem[w][tr];
    act8[r] = s_act[w][tr];
  }

  // A fragment: query for t = tbase + tl.
  // 32-bit A 16x4 layout: v0 = K0 (lanes 0-15) / K2 (16-31); v1 = K1 / K3.
  float qv[kD];
  {
    const int curq = s_cur[w][tl];
    const float remq = s_rem[w][tl];
    float c0 = 0.f, c1 = 0.f, c2 = 0.f, c3 = 0.f;
    if (curq != 0) {
      const float* pp = psi + ((size_t)b * kNP1 + curq) * kD;
      c0 = pp[0]; c1 = pp[1]; c2 = pp[2]; c3 = pp[3];
    }
    const float ctxv[kCTX] = {
        c0, c1, c2, c3,
        remq / cap,
        (float)(tbase + tl) / 200.0f,  // t_norm, n_customers = 200
        coords[((size_t)b * kNP1 + curq) * 2 + 0],
        coords[((size_t)b * kNP1 + curq) * 2 + 1]};
#pragma unroll
    for (int d0 = 0; d0 < kD; ++d0) {
      float acc = wqb[d0];
#pragma unroll
      for (int j = 0; j < kCTX; ++j) acc += wqw[d0][j] * ctxv[j];
      qv[d0] = acc;
    }
  }
  v2f afrag;
  afrag[0] = (lane < 16) ? qv[0] : qv[2];
  afrag[1] = (lane < 16) ? qv[1] : qv[3];

  // Tile loop over n: WMMA + per-element score terms + masking.
  float sc[kNTiles][8];
  unsigned hasC = 0;  // bit r: exists unmasked customer n>=1 in row r (half-local)
#pragma unroll
  for (int j = 0; j < kNTiles; ++j) {
    const int n = j * 16 + tl;
    const bool valid = (n < kNP1);
    float p0 = 0.f, p1 = 0.f, p2 = 0.f, p3 = 0.f;
    float demn = 0.f, nx = 0.f, ny = 0.f, itf = 0.f;
    if (valid) {
      const float* pp = psi + ((size_t)b * kNP1 + n) * kD;
      p0 = pp[0]; p1 = pp[1]; p2 = pp[2]; p3 = pp[3];
      demn = demands[(size_t)b * kNP1 + n];
      nx = coords[((size_t)b * kNP1 + n) * 2 + 0];
      ny = coords[((size_t)b * kNP1 + n) * 2 + 1];
      itf = interf_ws[(size_t)b * kNP1 + n];
    }
    // B 4x16 layout (mirror of A): v0 = K0 / K2 rows; v1 = K1 / K3, N = lane&15
    v2f bfrag;
    bfrag[0] = (lane < 16) ? p0 : p2;
    bfrag[1] = (lane < 16) ? p1 : p3;
    v8f cacc = {};
    cacc = __builtin_amdgcn_wmma_f32_16x16x4_f32(
        false, afrag, false, bfrag, (short)0, cacc, false, false);
#pragma unroll
    for (int r = 0; r < 8; ++r) {
      float s = cacc[r];
      const float dx = nx - curx[r];
      const float dy = ny - cury[r];
      const float dist = sqrtf(dx * dx + dy * dy + 1e-12f);
      s += lam * itf - mu * dist + nu * (demn / cap);
      const unsigned vw = s_vis[w][r + hi16 * 8][n >> 5];
      const bool visb = (vw >> (n & 31)) & 1u;
      const bool exc = demn > rem8[r];
      if (valid && (n >= 1) && !(visb || exc)) hasC |= (1u << r);
      if ((n >= 1) && (visb || exc)) s = -1.0e9f;
      if (!valid) s = -1.0e30f;
      sc[j][r] = s;
    }
  }

  // has_cust OR-reduction within each 16-lane half; fix up depot column n==0.
  int hc = (int)hasC;
  hc |= shflx_i(hc, 1); hc |= shflx_i(hc, 2);
  hc |= shflx_i(hc, 4); hc |= shflx_i(hc, 8);
  if (tl == 0) {
#pragma unroll
    for (int r = 0; r < 8; ++r)
      if (dep8[r] && ((hc >> r) & 1)) sc[0][r] = -1.0e9f;
  }

  // Row-wise log-softmax over 201 (+ padding at -1e30) and entropy.
  float logZ[8];
  float entHalf = 0.f;
#pragma unroll
  for (int r = 0; r < 8; ++r) {
    float m = -3.0e38f;
#pragma unroll
    for (int j = 0; j < kNTiles; ++j) m = fmaxf(m, sc[j][r]);
    m = fmaxf(m, shflx_f(m, 1)); m = fmaxf(m, shflx_f(m, 2));
    m = fmaxf(m, shflx_f(m, 4)); m = fmaxf(m, shflx_f(m, 8));
    float se = 0.f;
#pragma unroll
    for (int j = 0; j < kNTiles; ++j) se += __expf(sc[j][r] - m);
    se += shflx_f(se, 1); se += shflx_f(se, 2);
    se += shflx_f(se, 4); se += shflx_f(se, 8);
    const float lz = m + __logf(se);
    logZ[r] = lz;
    float e = 0.f;
#pragma unroll
    for (int j = 0; j < kNTiles; ++j) {
      const float lp = sc[j][r] - lz;
      e -= __expf(lp) * lp;
    }
    e += shflx_f(e, 1); e += shflx_f(e, 2);
    e += shflx_f(e, 4); e += shflx_f(e, 8);
    entHalf += e;
  }

  // Gather lp at actions: exactly one (lane, tile) matches per row.
#pragma unroll
  for (int j = 0; j < kNTiles; ++j) {
    const int n = j * 16 + tl;
#pragma unroll
    for (int r = 0; r < 8; ++r) {
      if (n == act8[r]) {
        out_lp[(size_t)b * kT + tbase + r + hi16 * 8] = sc[j][r] - logZ[r];
      }
    }
  }

  // Deterministic entropy partial per (b, t-tile).
  const float entTile = entHalf + shflx_f(entHalf, 16);
  if (lane == 0) ent_ws[b * 16 + ttile] = entTile;
}

// ---------------------------------------------------------------------------
// Kernel 4: entropy[b] = mean over T of per-t entropies (deterministic).
// ---------------------------------------------------------------------------
__global__ void ent_reduce(const float* __restrict__ ent_ws,
                           float* __restrict__ out_ent) {
  const int b = blockIdx.x * blockDim.x + threadIdx.x;
  if (b >= kMB) return;
  float s = 0.f;
#pragma unroll
  for (int i = 0; i < 16; ++i) s += ent_ws[b * 16 + i];
  out_ent[b] = s / (float)kT;
}

}  // namespace

extern "C" void kernel_launch(void* const* d_in, const int* in_sizes, int n_in,
                              void* d_out, int out_size, void* d_ws,
                              size_t ws_size, hipStream_t stream) {
  const float* demands = (const float*)d_in[0];
  const float* coords  = (const float*)d_in[1];
  const float* psi     = (const float*)d_in[2];
  const float* Wq_w    = (const float*)d_in[3];
  const float* Wq_b    = (const float*)d_in[4];
  const float* lam_p   = (const float*)d_in[5];
  const float* mu_p    = (const float*)d_in[6];
  const float* nu_p    = (const float*)d_in[7];
  const int*   actions = (const int*)d_in[8];
  const int*   knn     = (const int*)d_in[9];
  const int*   cap_p   = (const int*)d_in[10];

  char* ws = (char*)d_ws;
  size_t off = 0;
  int* cur_ws = (int*)(ws + off);          off += (size_t)kMB * kT * 4;
  float* used_ws = (float*)(ws + off);     off += (size_t)kMB * kT * 4;
  unsigned* vis_ws = (unsigned*)(ws + off); off += (size_t)kMB * kT * 7 * 4;
  float* interf_ws = (float*)(ws + off);   off += (size_t)kMB * kNP1 * 4;
  float* ent_ws = (float*)(ws + off);      off += (size_t)kMB * 16 * 4;

  float* out_lp = (float*)d_out;            // [MB*T]
  float* out_ent = out_lp + (size_t)kMB * kT;  // [MB]

  scan_kernel<<<(kMB + 255) / 256, 256, 0, stream>>>(demands, actions, cur_ws,
                                                     used_ws, vis_ws);
  interf_kernel<<<(kMB * kNP1 + 255) / 256, 256, 0, stream>>>(psi, knn,
                                                              interf_ws);
  policy_main<<<kMB * 2, 256, 0, stream>>>(demands, coords, psi, Wq_w, Wq_b,
                                           lam_p, mu_p, nu_p, actions, cap_p,
                                           cur_ws, used_ws, vis_ws, interf_ws,
                                           out_lp, ent_ws);
  ent_reduce<<<(kMB + 255) / 256, 256, 0, stream>>>(ent_ws, out_ent);
}